// MPMModel_84069689852237
// MI455X (gfx1250) — compile-verified
//
#include <hip/hip_runtime.h>
#include <math.h>

typedef float vfloat2 __attribute__((ext_vector_type(2)));
typedef float vfloat4 __attribute__((ext_vector_type(4)));

namespace {
constexpr int   GSZ    = 256;
constexpr float DX     = 1.0f / 256.0f;
constexpr float INV_DX = 256.0f;
constexpr float DT     = 1e-4f;
constexpr float P_VOL  = (DX * 0.5f) * (DX * 0.5f);
constexpr float P_MASS = P_VOL * 1.0f;
constexpr float MU0    = 1000.0f / (2.0f * (1.0f + 0.2f));                  // E/(2(1+nu))
constexpr float LAM0   = 1000.0f * 0.2f / ((1.0f + 0.2f) * (1.0f - 0.4f)); // E*nu/((1+nu)(1-2nu))
constexpr float STRESS_COEF = -DT * P_VOL * 4.0f * INV_DX * INV_DX;        // == -DT
constexpr int   CELL_STRIDE = 4;   // mvx, mvy, mass, pad  (16B cells)
}

// ---------------------------------------------------------------- zero grid (vector stores)
__global__ void mpm_zero_kernel(vfloat4* __restrict__ g, int n4) {
  int i = blockIdx.x * blockDim.x + threadIdx.x;
  if (i < n4) g[i] = (vfloat4)0.0f;
}

// ---------------------------------------------------------------- P2G + material update
__global__ void mpm_p2g_kernel(const float* __restrict__ x, const float* __restrict__ v,
                               const float* __restrict__ Cin, const float* __restrict__ Fin,
                               const int* __restrict__ mat, const float* __restrict__ Jpin,
                               float* __restrict__ grid, float* __restrict__ out, int n) {
  int i = blockIdx.x * blockDim.x + threadIdx.x;
  if (i >= n) return;

  // x is re-read in G2P -> regular (L2-resident) load. One-shot streams -> NT loads.
  vfloat2 xp = ((const vfloat2*)x)[i];
  float px = xp.x, py = xp.y;
  float xg = px * INV_DX, yg = py * INV_DX;
  float bxf = floorf(xg - 0.5f), byf = floorf(yg - 0.5f);
  int   bx = (int)bxf, by = (int)byf;
  float fx = xg - bxf, fy = yg - byf;

  float wx[3], wy[3];
  wx[0] = 0.5f * (1.5f - fx) * (1.5f - fx);
  wx[1] = 0.75f - (fx - 1.0f) * (fx - 1.0f);
  wx[2] = 0.5f * (fx - 0.5f) * (fx - 0.5f);
  wy[0] = 0.5f * (1.5f - fy) * (1.5f - fy);
  wy[1] = 0.75f - (fy - 1.0f) * (fy - 1.0f);
  wy[2] = 0.5f * (fy - 0.5f) * (fy - 0.5f);

  vfloat4 Cv = __builtin_nontemporal_load(&((const vfloat4*)Cin)[i]);
  vfloat4 Fv = __builtin_nontemporal_load(&((const vfloat4*)Fin)[i]);
  float C00 = Cv.x, C01 = Cv.y, C10 = Cv.z, C11 = Cv.w;
  float F00 = Fv.x, F01 = Fv.y, F10 = Fv.z, F11 = Fv.w;

  // F <- F + DT * (C @ F)
  float nF00 = F00 + DT * (C00 * F00 + C01 * F10);
  float nF01 = F01 + DT * (C00 * F01 + C01 * F11);
  float nF10 = F10 + DT * (C10 * F00 + C11 * F10);
  float nF11 = F11 + DT * (C10 * F01 + C11 * F11);

  int   m  = __builtin_nontemporal_load(&mat[i]);
  float jp = __builtin_nontemporal_load(&Jpin[i]);

  float h   = (m == 1) ? 0.3f : expf(10.0f * (1.0f - jp));
  float mu  = (m == 0) ? 0.0f : MU0 * h;
  float lam = LAM0 * h;

  // Closed-form 2x2 SVD: nF = R(u) * diag(s0,s1) * R(w), u=(a2+a1)/2, w=(a2-a1)/2
  float E  = 0.5f * (nF00 + nF11), Fh = 0.5f * (nF00 - nF11);
  float Gh = 0.5f * (nF10 + nF01), Hh = 0.5f * (nF10 - nF01);
  float q  = sqrtf(E * E + Hh * Hh);
  float r  = sqrtf(Fh * Fh + Gh * Gh);
  float s0 = q + r, s1 = q - r;                 // descending; s1>0 since det(F)~1
  float a1 = atan2f(Gh, Fh), a2 = atan2f(Hh, E);
  float thU = 0.5f * (a2 + a1), thV = 0.5f * (a2 - a1);
  float cu, su, cvt, svt;
  sincosf(thU, &su, &cu);
  sincosf(thV, &svt, &cvt);

  bool snow = (m == 2);
  float c0 = fminf(fmaxf(s0, 1.0f - 2.5e-2f), 1.0f + 4.5e-3f);
  float c1 = fminf(fmaxf(s1, 1.0f - 2.5e-2f), 1.0f + 4.5e-3f);
  if (snow) { jp *= (s0 / c0) * (s1 / c1); s0 = c0; s1 = c1; }
  float J = s0 * s1;

  float rF00, rF01, rF10, rF11;
  if (m == 0) {                                  // liquid: F = sqrt(J) * I
    float s = sqrtf(J);
    rF00 = s; rF01 = 0.0f; rF10 = 0.0f; rF11 = s;
  } else if (snow) {                             // F = U diag(s) Vh
    rF00 =  cu * s0 * cvt - su * s1 * svt;
    rF01 = -cu * s0 * svt - su * s1 * cvt;
    rF10 =  su * s0 * cvt + cu * s1 * svt;
    rF11 = -su * s0 * svt + cu * s1 * cvt;
  } else {
    rF00 = nF00; rF01 = nF01; rF10 = nF10; rF11 = nF11;
  }

  // R = U @ Vh = R(u+w)
  float cr = cu * cvt - su * svt;
  float sr = su * cvt + cu * svt;
  // stress = coef * (2 mu (F-R) F^T + lam J (J-1) I)
  float A00 = rF00 - cr, A01 = rF01 + sr, A10 = rF10 - sr, A11 = rF11 - cr;
  float S00 = A00 * rF00 + A01 * rF01;
  float S01 = A00 * rF10 + A01 * rF11;
  float S10 = A10 * rF00 + A11 * rF01;
  float S11 = A10 * rF10 + A11 * rF11;
  float twoMu = 2.0f * mu;
  float lj = lam * J * (J - 1.0f);
  S00 = STRESS_COEF * (twoMu * S00 + lj);
  S01 = STRESS_COEF * (twoMu * S01);
  S10 = STRESS_COEF * (twoMu * S10);
  S11 = STRESS_COEF * (twoMu * S11 + lj);

  float aff00 = S00 + P_MASS * C00, aff01 = S01 + P_MASS * C01;
  float aff10 = S10 + P_MASS * C10, aff11 = S11 + P_MASS * C11;

  vfloat2 vp = __builtin_nontemporal_load(&((const vfloat2*)v)[i]);
  float mvx = P_MASS * vp.x, mvy = P_MASS * vp.y;

  // Fused scatter: w * (P_MASS*v + aff @ dpos), dpos = node_pos - x_p = ((ii-fx)DX, (jj-fy)DX)
  // 3 channels instead of 7 -> 27 L2-resident atomics per particle (vs 63)
#pragma unroll
  for (int ii = 0; ii < 3; ++ii) {
    float dpx = ((float)ii - fx) * DX;
#pragma unroll
    for (int jj = 0; jj < 3; ++jj) {
      float dpy = ((float)jj - fy) * DX;
      float wgt = wx[ii] * wy[jj];
      float* cell = grid + (size_t)((bx + ii) * GSZ + (by + jj)) * CELL_STRIDE;
      atomicAdd(cell + 0, wgt * (mvx + aff00 * dpx + aff01 * dpy));
      atomicAdd(cell + 1, wgt * (mvy + aff10 * dpx + aff11 * dpy));
      atomicAdd(cell + 2, wgt * P_MASS);
    }
  }

  size_t N = (size_t)n;
  vfloat4 outF; outF.x = rF00; outF.y = rF01; outF.z = rF10; outF.w = rF11;
  __builtin_nontemporal_store(outF, &((vfloat4*)(out + 8 * N))[i]);
  __builtin_nontemporal_store((float)m, &out[12 * N + i]);
  __builtin_nontemporal_store(jp, &out[13 * N + i]);
}

// ---------------------------------------------------------------- grid momentum -> velocity
__global__ void mpm_grid_kernel(const vfloat4* __restrict__ grid, vfloat2* __restrict__ gridv) {
  int idx = blockIdx.x * blockDim.x + threadIdx.x;
  if (idx >= GSZ * GSZ) return;
  vfloat4 cell = grid[idx];
  float msum = cell.z;
  int gi = idx >> 8, gj = idx & (GSZ - 1);
  float denom = (msum > 0.0f) ? msum : 1.0f;
  float vx = cell.x / denom;
  float vy = cell.y / denom;
  vy -= DT * 50.0f;
  if (gi < 3)        vx = fmaxf(vx, 0.0f);
  if (gi >= GSZ - 2) vx = fminf(vx, 0.0f);
  if (gj < 3)        vy = fmaxf(vy, 0.0f);
  if (gj >= GSZ - 2) vy = fminf(vy, 0.0f);
  vfloat2 r; r.x = vx; r.y = vy;
  gridv[idx] = r;
}

// ---------------------------------------------------------------- G2P gather + advect
__global__ void mpm_g2p_kernel(const float* __restrict__ x, const vfloat2* __restrict__ gridv,
                               float* __restrict__ out, int n) {
  int i = blockIdx.x * blockDim.x + threadIdx.x;
  if (i >= n) return;
  vfloat2 xp = ((const vfloat2*)x)[i];
  float px = xp.x, py = xp.y;
  float xg = px * INV_DX, yg = py * INV_DX;
  float bxf = floorf(xg - 0.5f), byf = floorf(yg - 0.5f);
  int   bx = (int)bxf, by = (int)byf;
  float fx = xg - bxf, fy = yg - byf;

  float wx[3], wy[3];
  wx[0] = 0.5f * (1.5f - fx) * (1.5f - fx);
  wx[1] = 0.75f - (fx - 1.0f) * (fx - 1.0f);
  wx[2] = 0.5f * (fx - 0.5f) * (fx - 0.5f);
  wy[0] = 0.5f * (1.5f - fy) * (1.5f - fy);
  wy[1] = 0.75f - (fy - 1.0f) * (fy - 1.0f);
  wy[2] = 0.5f * (fy - 0.5f) * (fy - 0.5f);

  float nvx = 0.0f, nvy = 0.0f;
  float ct00 = 0.0f, ct01 = 0.0f, ct10 = 0.0f, ct11 = 0.0f;
#pragma unroll
  for (int ii = 0; ii < 3; ++ii) {
#pragma unroll
    for (int jj = 0; jj < 3; ++jj) {
      int gi = bx + ii, gj = by + jj;
      float wgt = wx[ii] * wy[jj];
      vfloat2 gv = gridv[gi * GSZ + gj];      // 64-bit load, L2-resident
      nvx += wgt * gv.x;
      nvy += wgt * gv.y;
      float qx = gi * DX, qy = gj * DX;
      ct00 += wgt * gv.x * qx;
      ct01 += wgt * gv.x * qy;
      ct10 += wgt * gv.y * qx;
      ct11 += wgt * gv.y * qy;
    }
  }
  float k4 = 4.0f * INV_DX * INV_DX;
  vfloat4 Cout;
  Cout.x = k4 * (ct00 - nvx * px);
  Cout.y = k4 * (ct01 - nvx * py);
  Cout.z = k4 * (ct10 - nvy * px);
  Cout.w = k4 * (ct11 - nvy * py);

  size_t N = (size_t)n;
  vfloat2 xout; xout.x = px + DT * nvx; xout.y = py + DT * nvy;
  vfloat2 vout; vout.x = nvx;           vout.y = nvy;
  __builtin_nontemporal_store(xout, &((vfloat2*)out)[i]);
  __builtin_nontemporal_store(vout, &((vfloat2*)(out + 2 * N))[i]);
  __builtin_nontemporal_store(Cout, &((vfloat4*)(out + 4 * N))[i]);
}

// ---------------------------------------------------------------- launch
extern "C" void kernel_launch(void* const* d_in, const int* in_sizes, int n_in,
                              void* d_out, int out_size, void* d_ws, size_t ws_size,
                              hipStream_t stream) {
  const float* x  = (const float*)d_in[0];
  const float* v  = (const float*)d_in[1];
  const float* C  = (const float*)d_in[2];
  const float* F  = (const float*)d_in[3];
  const int*   mt = (const int*)d_in[4];
  const float* Jp = (const float*)d_in[5];
  const int n = in_sizes[4];   // N particles (material count)

  float*   out   = (float*)d_out;
  float*   grid  = (float*)d_ws;                            // GSZ*GSZ*4 floats (1 MB)
  vfloat2* gridv = (vfloat2*)(grid + (size_t)GSZ * GSZ * CELL_STRIDE); // 512 KB

  const int BLK = 256;  // 8 wave32 waves
  int nCells = GSZ * GSZ;          // each cell is one vfloat4
  hipLaunchKernelGGL(mpm_zero_kernel, dim3((nCells + BLK - 1) / BLK), dim3(BLK), 0, stream,
                     (vfloat4*)grid, nCells);
  hipLaunchKernelGGL(mpm_p2g_kernel, dim3((n + BLK - 1) / BLK), dim3(BLK), 0, stream,
                     x, v, C, F, mt, Jp, grid, out, n);
  hipLaunchKernelGGL(mpm_grid_kernel, dim3((nCells + BLK - 1) / BLK), dim3(BLK), 0, stream,
                     (const vfloat4*)grid, gridv);
  hipLaunchKernelGGL(mpm_g2p_kernel, dim3((n + BLK - 1) / BLK), dim3(BLK), 0, stream,
                     x, gridv, out, n);
}